// GCNLayer_42657615184064
// MI455X (gfx1250) — compile-verified
//
#include <hip/hip_runtime.h>
#include <hip/hip_bf16.h>

#define N_NODES 100000
#define N_EDGES 1600000
#define D 128

typedef __attribute__((ext_vector_type(2))) float v2f;
typedef __attribute__((ext_vector_type(8))) float v8f;

// ---------------- degree ----------------
__global__ void k_zero_deg(unsigned* __restrict__ deg) {
    int i = blockIdx.x * blockDim.x + threadIdx.x;
    if (i < N_NODES) deg[i] = 0u;
}

__global__ void k_count_deg(const long long* __restrict__ ei, unsigned* __restrict__ deg) {
    int e = blockIdx.x * blockDim.x + threadIdx.x;
    if (e < N_EDGES) {
        int d = (int)ei[N_EDGES + e];   // dst row
        atomicAdd(&deg[d], 1u);
    }
}

__global__ void k_dinv(const unsigned* __restrict__ deg, float* __restrict__ dinv) {
    int i = blockIdx.x * blockDim.x + threadIdx.x;
    if (i < N_NODES) {
        float d = (float)deg[i] + 1.0f;   // +1 self loop; always > 0
        dinv[i] = rsqrtf(d);
    }
}

// ---------------- GEMM h = x @ W^T via V_WMMA_F32_16X16X4_F32 ----------------
// Block: 256 threads = 8 waves. Block b owns rows [16b, 16b+16); wave w owns cols [16w, 16w+16).
// f32 A (16x4) frag layout: lanes 0-15 -> M=lane, VGPR0=K0,VGPR1=K1; lanes 16-31 -> K2,K3.
// f32 B (4x16) frag layout mirrors A with lanes holding N.
__global__ void __launch_bounds__(256) k_gemm_wmma(const float* __restrict__ x,
                                                  const float* __restrict__ W,
                                                  float* __restrict__ h) {
    const int lane = threadIdx.x & 31;
    const int wave = threadIdx.x >> 5;
    const int m0 = blockIdx.x * 16;
    const int n0 = wave * 16;

    const int arow  = m0 + (lane & 15);        // A: row of x
    const int bcol  = n0 + (lane & 15);        // B: col of h = row of W
    const int khalf = (lane >> 4) * 2;         // 0 for lanes 0-15, 2 for lanes 16-31

    const float* ap = x + (size_t)arow * D + khalf;
    const float* bp = W + (size_t)bcol * D + khalf;

    v8f c = {};
#pragma unroll
    for (int k0 = 0; k0 < D; k0 += 4) {
        v2f a = *(const v2f*)(ap + k0);
        v2f b = *(const v2f*)(bp + k0);
        c = __builtin_amdgcn_wmma_f32_16x16x4_f32(
            /*neg_a=*/false, a, /*neg_b=*/false, b,
            /*c_mod=*/(short)0, c, /*reuse_a=*/false, /*reuse_b=*/false);
    }

    // C/D 16x16 f32 layout: VGPR v holds row m0 + v + (lane>=16 ? 8 : 0), col = bcol.
    const int rbase = m0 + (lane >> 4) * 8;
#pragma unroll
    for (int v = 0; v < 8; ++v) {
        h[(size_t)(rbase + v) * D + bcol] = c[v];
    }
}

// ---------------- self-loop init: out[n] = h[n] * dinv[n]^2 ----------------
__global__ void k_self_loop(const float* __restrict__ h, const float* __restrict__ dinv,
                            float* __restrict__ out) {
    long long i = (long long)blockIdx.x * blockDim.x + threadIdx.x;  // one float4 per thread
    int n  = (int)(i >> 5);
    int c4 = ((int)i & 31) * 4;
    if (n < N_NODES) {
        float w = dinv[n] * dinv[n];
        float4 v = *(const float4*)(h + (size_t)n * D + c4);
        float4 r = make_float4(v.x * w, v.y * w, v.z * w, v.w * w);
        *(float4*)(out + (size_t)n * D + c4) = r;
    }
}

// ---------------- edge scatter: 1 wave per edge ----------------
__global__ void __launch_bounds__(256) k_scatter(const long long* __restrict__ ei,
                                                 const float* __restrict__ h,
                                                 const float* __restrict__ dinv,
                                                 float* __restrict__ out) {
    const int lane = threadIdx.x & 31;
    const long long e = (long long)blockIdx.x * 8 + (threadIdx.x >> 5);
    if (e >= N_EDGES) return;
    const int s = (int)ei[e];
    const int d = (int)ei[N_EDGES + e];
    const float w = dinv[s] * dinv[d];
    const float4 v = *(const float4*)(h + (size_t)s * D + lane * 4);
    float* o = out + (size_t)d * D + lane * 4;
    unsafeAtomicAdd(o + 0, v.x * w);
    unsafeAtomicAdd(o + 1, v.y * w);
    unsafeAtomicAdd(o + 2, v.z * w);
    unsafeAtomicAdd(o + 3, v.w * w);
}

// ---------------- epilogue: out = relu(out + b) in place ----------------
__global__ void k_bias_relu(float* __restrict__ out, const float* __restrict__ b) {
    long long i = (long long)blockIdx.x * blockDim.x + threadIdx.x;
    int n  = (int)(i >> 5);
    int c4 = ((int)i & 31) * 4;
    if (n < N_NODES) {
        float4 bb = *(const float4*)(b + c4);
        float* p = out + (size_t)n * D + c4;
        float4 v = *(const float4*)p;
        v.x = fmaxf(v.x + bb.x, 0.0f);
        v.y = fmaxf(v.y + bb.y, 0.0f);
        v.z = fmaxf(v.z + bb.z, 0.0f);
        v.w = fmaxf(v.w + bb.w, 0.0f);
        *(float4*)p = v;
    }
}

extern "C" void kernel_launch(void* const* d_in, const int* in_sizes, int n_in,
                              void* d_out, int out_size, void* d_ws, size_t ws_size,
                              hipStream_t stream) {
    const float*     x  = (const float*)d_in[0];
    const long long* ei = (const long long*)d_in[1];   // int64 [2, E]
    const float*     W  = (const float*)d_in[2];       // [D_OUT, D_IN] row-major
    const float*     b  = (const float*)d_in[3];
    float* out = (float*)d_out;

    char* ws = (char*)d_ws;
    float*    h    = (float*)ws;                              // 51,200,000 B
    unsigned* deg  = (unsigned*)(ws + (size_t)51200000);      //    400,000 B
    float*    dinv = (float*)(ws + (size_t)51600000);         //    400,000 B

    k_zero_deg <<<(N_NODES + 255) / 256, 256, 0, stream>>>(deg);
    k_count_deg<<<(N_EDGES + 255) / 256, 256, 0, stream>>>(ei, deg);
    k_dinv     <<<(N_NODES + 255) / 256, 256, 0, stream>>>(deg, dinv);

    k_gemm_wmma<<<N_NODES / 16, 256, 0, stream>>>(x, W, h);   // 6250 blocks, no tail

    long long nd4 = (long long)N_NODES * 32;                  // one float4 per thread
    k_self_loop<<<(unsigned)((nd4 + 255) / 256), 256, 0, stream>>>(h, dinv, out);
    k_scatter  <<<(N_EDGES + 7) / 8, 256, 0, stream>>>(ei, h, dinv, out);
    k_bias_relu<<<(unsigned)((nd4 + 255) / 256), 256, 0, stream>>>(out, b);
}